// VectorQuantizer_11536282157447
// MI455X (gfx1250) — compile-verified
//
#include <hip/hip_runtime.h>

#define DIM 64
#define NEMB 512

typedef __attribute__((ext_vector_type(2))) float v2f;
typedef __attribute__((ext_vector_type(8))) float v8f;

// Pre-kernel: enorm[k] = sum_d E[d][k]^2   (E is row-major [DIM][NEMB])
__global__ void vq_enorm_kernel(const float* __restrict__ E, float* __restrict__ enorm) {
  int k = blockIdx.x * blockDim.x + threadIdx.x;
  if (k < NEMB) {
    float s = 0.f;
    #pragma unroll
    for (int d = 0; d < DIM; ++d) { float v = E[d * NEMB + k]; s += v * v; }
    enorm[k] = s;
  }
}

// Main: per 64-row block, score = enorm[k] - 2 * x.e_k via f32 WMMA, argmin, gather.
__global__ __launch_bounds__(256)
void vq_main_kernel(const float* __restrict__ X, const float* __restrict__ E,
                    const float* __restrict__ enorm,
                    float* __restrict__ outq, float* __restrict__ outi) {
  __shared__ float lds_x[64 * 68];     // 64 rows, stride 68 floats (pad: conflict-free)
  __shared__ float red_val[8][64];
  __shared__ int   red_idx[8][64];
  __shared__ int   best_s[64];

  const int tid  = threadIdx.x;
  const int lane = tid & 31;
  const int wave = tid >> 5;
  const int n    = lane & 15;          // column within 16-wide tile
  const int g    = lane >> 4;          // half-wave group
  const long row0 = (long)blockIdx.x * 64;

  // ---- stage X[row0:row0+64, 0:64] into LDS (coalesced float4) ----
  const float4* xin = (const float4*)(X + row0 * DIM);
  #pragma unroll
  for (int i = 0; i < 4; ++i) {
    int e = tid + i * 256;             // 1024 float4 total
    int r = e >> 4, c = e & 15;
    float4 v = xin[e];
    *(float4*)&lds_x[r * 68 + c * 4] = v;
  }
  __syncthreads();

  const int cbase = wave * 64;         // this wave's 64-column slice of K

  v8f acc[4][4];
  #pragma unroll
  for (int rt = 0; rt < 4; ++rt)
    #pragma unroll
    for (int t = 0; t < 4; ++t)
      acc[rt][t] = (v8f){0.f, 0.f, 0.f, 0.f, 0.f, 0.f, 0.f, 0.f};

  // ---- GEMM: 16 K-steps of 4, 4 row tiles x 4 col tiles of 16x16 ----
  #pragma unroll
  for (int s = 0; s < 16; ++s) {
    const int d0 = 4 * s + 2 * g;      // A/B frag layout: lane = idx + 16*(k>>1), vgpr = k&1
    v2f bfr[4];
    #pragma unroll
    for (int t = 0; t < 4; ++t) {
      int col = cbase + t * 16 + n;
      bfr[t].x = E[d0 * NEMB + col];
      bfr[t].y = E[(d0 + 1) * NEMB + col];
    }
    #pragma unroll
    for (int rt = 0; rt < 4; ++rt) {
      v2f afr = *(const v2f*)&lds_x[(rt * 16 + n) * 68 + d0];
      #pragma unroll
      for (int t = 0; t < 4; ++t)
        acc[rt][t] = __builtin_amdgcn_wmma_f32_16x16x4_f32(
            false, afr, false, bfr[t], (short)0, acc[rt][t], false, false);
    }
  }

  // ---- scoring + argmin ----
  float en[4];
  #pragma unroll
  for (int t = 0; t < 4; ++t) en[t] = enorm[cbase + t * 16 + n];

  #pragma unroll
  for (int rt = 0; rt < 4; ++rt) {
    float bv[8]; int bi[8];
    #pragma unroll
    for (int r = 0; r < 8; ++r) {
      // C/D layout: vgpr r, lanes 0-15 -> row r, lanes 16-31 -> row r+8, col = lane&15
      float best = 3.4e38f; int bidx = NEMB;
      #pragma unroll
      for (int t = 0; t < 4; ++t) {
        float sc  = en[t] - 2.0f * acc[rt][t][r];
        int   idx = cbase + t * 16 + n;
        if (sc < best || (sc == best && idx < bidx)) { best = sc; bidx = idx; }
      }
      // butterfly min-reduce across the 16 lanes sharing these rows
      #pragma unroll
      for (int off = 8; off >= 1; off >>= 1) {
        float ov = __shfl_xor(best, off, 16);
        int   oi = __shfl_xor(bidx, off, 16);
        if (ov < best || (ov == best && oi < bidx)) { best = ov; bidx = oi; }
      }
      bv[r] = best; bi[r] = bidx;
    }
    if (n == 0) {
      #pragma unroll
      for (int r = 0; r < 8; ++r) {
        red_val[wave][rt * 16 + g * 8 + r] = bv[r];
        red_idx[wave][rt * 16 + g * 8 + r] = bi[r];
      }
    }
  }
  __syncthreads();

  // ---- cross-wave argmin (first-min tie-break: lower index) ----
  if (tid < 64) {
    float best = red_val[0][tid]; int bidx = red_idx[0][tid];
    #pragma unroll
    for (int w = 1; w < 8; ++w) {
      float v = red_val[w][tid]; int i2 = red_idx[w][tid];
      if (v < best || (v == best && i2 < bidx)) { best = v; bidx = i2; }
    }
    best_s[tid] = bidx;
    outi[row0 + tid] = (float)bidx;
  }
  __syncthreads();

  // ---- gather-write quantized rows (coalesced stores; E gathers hit L2) ----
  #pragma unroll
  for (int i = 0; i < 16; ++i) {
    int e = tid + i * 256;             // 4096 floats
    int r = e >> 6, d = e & 63;
    outq[(row0 + r) * DIM + d] = E[d * NEMB + best_s[r]];
  }
}

extern "C" void kernel_launch(void* const* d_in, const int* in_sizes, int n_in,
                              void* d_out, int out_size, void* d_ws, size_t ws_size,
                              hipStream_t stream) {
  (void)n_in; (void)out_size; (void)ws_size;
  const float* X = (const float*)d_in[0];   // [N, 64] f32 (flattened 64x32x32x64)
  const float* E = (const float*)d_in[1];   // [64, 512] f32
  float* enorm = (float*)d_ws;              // 512 floats scratch
  float* outq  = (float*)d_out;             // quantized [N,64] then indices [N]
  const int N = in_sizes[0] / DIM;
  float* outi = outq + (size_t)N * DIM;

  vq_enorm_kernel<<<(NEMB + 255) / 256, 256, 0, stream>>>(E, enorm);
  vq_main_kernel<<<N / 64, 256, 0, stream>>>(X, E, enorm, outq, outi);
}